// NodeEmbedding_9234179687481
// MI455X (gfx1250) — compile-verified
//
#include <hip/hip_runtime.h>

typedef __attribute__((ext_vector_type(2))) float v2f;
typedef __attribute__((ext_vector_type(8))) float v8f;

// ---------------------------------------------------------------------------
// 1) deg[i] = 1.0f  (self loop)
// ---------------------------------------------------------------------------
__global__ void gcn_init_deg(float* __restrict__ deg, int N) {
    int i = blockIdx.x * blockDim.x + threadIdx.x;
    if (i < N) deg[i] = 1.0f;
}

// ---------------------------------------------------------------------------
// 2) deg[dst[e]] += 1
// ---------------------------------------------------------------------------
__global__ void gcn_accum_deg(const int* __restrict__ ei, float* __restrict__ deg,
                              int E) {
    int e = blockIdx.x * blockDim.x + threadIdx.x;
    if (e < E) atomicAdd(&deg[ei[E + e]], 1.0f);
}

// ---------------------------------------------------------------------------
// 3) dinv[i] = rsqrt(deg[i])   (deg >= 1 always due to self loops)
// ---------------------------------------------------------------------------
__global__ void gcn_deg_to_dinv(float* __restrict__ d, int N) {
    int i = blockIdx.x * blockDim.x + threadIdx.x;
    if (i < N) d[i] = __frsqrt_rn(d[i]);
}

// ---------------------------------------------------------------------------
// 4) h = x @ W   (f32 WMMA 16x16x4, K = 128 => 32 chained WMMAs per tile)
//    Block: 256 threads = 8 waves. Block handles 16 rows; wave w handles
//    columns [16w, 16w+16). B fragments preloaded to registers once and
//    reused across a grid-stride loop over row tiles.
// ---------------------------------------------------------------------------
__global__ __launch_bounds__(256) void gcn_gemm_wmma(const float* __restrict__ x,
                                                     const float* __restrict__ W,
                                                     float* __restrict__ h,
                                                     int nrows) {
    __shared__ float xs[16][132];                 // padded: stride 132 spreads banks

    const int lane = threadIdx.x & 31;
    const int wave = threadIdx.x >> 5;            // 0..7 -> column tile
    const int col0 = wave * 16;
    const int r    = lane & 15;                   // row (A) / col (B,C) within tile
    const int kh   = lane >> 4;                   // K-half select (ISA 32-bit layout)
    const int col  = col0 + r;

    // Preload ALL B fragments for this wave's 16 columns: K=0..127, 32 steps.
    // VGPR j of fragment s holds K = 4s + 2*kh + j (ISA §7.12.2 32-bit B layout).
    v2f bfrag[32];
#pragma unroll
    for (int s = 0; s < 32; ++s) {
        int k = s * 4 + 2 * kh;
        bfrag[s].x = W[(k + 0) * 128 + col];
        bfrag[s].y = W[(k + 1) * 128 + col];
    }

    const int nTiles = (nrows + 15) >> 4;
    for (int rt = blockIdx.x; rt < nTiles; rt += gridDim.x) {
        const int row0 = rt << 4;

        __syncthreads();                          // protect LDS from prior readers
        // Cooperative load of x tile [16 x 128]: 512 float4, 2 per thread.
#pragma unroll
        for (int i = 0; i < 2; ++i) {
            int f4  = threadIdx.x + i * 256;      // 0..511
            int rr  = f4 >> 5;                    // 32 float4 per row
            int cc  = (f4 & 31) * 4;
            int row = row0 + rr;
            float4 v = make_float4(0.f, 0.f, 0.f, 0.f);
            if (row < nrows)
                v = *(const float4*)(x + (size_t)row * 128 + cc);
            *(float4*)(&xs[rr][cc]) = v;
        }
        __syncthreads();

        v8f acc = {};
#pragma unroll
        for (int s = 0; s < 32; ++s) {
            int k = s * 4 + 2 * kh;
            v2f a;
            a.x = xs[r][k];                       // A: VGPR j holds K = 4s+2*kh+j
            a.y = xs[r][k + 1];
            acc = __builtin_amdgcn_wmma_f32_16x16x4_f32(
                /*neg_a=*/false, a, /*neg_b=*/false, bfrag[s],
                /*c_mod=*/(short)0, acc, /*reuse_a=*/false, /*reuse_b=*/false);
        }

        // C/D layout: VGPR i -> M = i + 8*kh, N = lane&15.
#pragma unroll
        for (int i = 0; i < 8; ++i) {
            int row = row0 + i + 8 * kh;
            if (row < nrows)
                h[(size_t)row * 128 + col] = acc[i];
        }
    }
}

// ---------------------------------------------------------------------------
// 5) out[i,:] = dinv[i]^2 * h[i,:]   (self-loop term; also overwrites poison)
// ---------------------------------------------------------------------------
__global__ void gcn_self_init(const float* __restrict__ h,
                              const float* __restrict__ dinv,
                              float* __restrict__ out, int N) {
    int t = blockIdx.x * blockDim.x + threadIdx.x;     // N*32 float4 slots
    if (t >= N * 32) return;
    int node = t >> 5;
    float s = dinv[node];
    s = s * s;
    float4 v = ((const float4*)h)[t];
    v.x *= s; v.y *= s; v.z *= s; v.w *= s;
    ((float4*)out)[t] = v;
}

// ---------------------------------------------------------------------------
// 6) edge scatter: one wave per edge, lane l handles channels 4l..4l+3.
//    out[dst,:] += dinv[src]*dinv[dst] * h[src,:]
// ---------------------------------------------------------------------------
__global__ __launch_bounds__(256) void gcn_edge_scatter(const int* __restrict__ ei,
                                                        const float* __restrict__ h,
                                                        const float* __restrict__ dinv,
                                                        float* __restrict__ out,
                                                        int E) {
    int wid  = (blockIdx.x * blockDim.x + threadIdx.x) >> 5;   // edge id
    int lane = threadIdx.x & 31;
    if (wid >= E) return;
    int src = ei[wid];
    int dst = ei[E + wid];
    float norm = dinv[src] * dinv[dst];

    float4 v = ((const float4*)(h + (size_t)src * 128))[lane];
    float* o = out + (size_t)dst * 128 + 4 * lane;
    atomicAdd(o + 0, v.x * norm);
    atomicAdd(o + 1, v.y * norm);
    atomicAdd(o + 2, v.z * norm);
    atomicAdd(o + 3, v.w * norm);
}

// ---------------------------------------------------------------------------
// 7) out = relu(out + b)
// ---------------------------------------------------------------------------
__global__ void gcn_finalize(float* __restrict__ out, const float* __restrict__ b,
                             int total4) {
    int t = blockIdx.x * blockDim.x + threadIdx.x;
    if (t >= total4) return;
    int c = (t & 31) * 4;                  // channel base within node
    float4 v = ((float4*)out)[t];
    v.x = fmaxf(v.x + b[c + 0], 0.f);
    v.y = fmaxf(v.y + b[c + 1], 0.f);
    v.z = fmaxf(v.z + b[c + 2], 0.f);
    v.w = fmaxf(v.w + b[c + 3], 0.f);
    ((float4*)out)[t] = v;
}

// ---------------------------------------------------------------------------
extern "C" void kernel_launch(void* const* d_in, const int* in_sizes, int n_in,
                              void* d_out, int out_size, void* d_ws, size_t ws_size,
                              hipStream_t stream) {
    const float* x  = (const float*)d_in[0];
    const int*   ei = (const int*)d_in[1];
    const float* W  = (const float*)d_in[2];
    const float* b  = (const float*)d_in[3];
    float* out = (float*)d_out;

    const int N = in_sizes[0] / 128;
    const int E = in_sizes[1] / 2;

    // Workspace layout: h [N,128] f32, then dinv [N] f32.
    float* h    = (float*)d_ws;
    float* dinv = h + (size_t)N * 128;

    const int B = 256;

    // 1-3: degrees -> dinv
    gcn_init_deg<<<(N + B - 1) / B, B, 0, stream>>>(dinv, N);
    gcn_accum_deg<<<(E + B - 1) / B, B, 0, stream>>>(ei, dinv, E);
    gcn_deg_to_dinv<<<(N + B - 1) / B, B, 0, stream>>>(dinv, N);

    // 4: h = x @ W (WMMA f32)
    int nTiles = (N + 15) / 16;
    int gemmBlocks = nTiles < 4096 ? nTiles : 4096;
    gcn_gemm_wmma<<<gemmBlocks, B, 0, stream>>>(x, W, h, N);

    // 5: self-loop term initializes out
    gcn_self_init<<<(N * 32 + B - 1) / B, B, 0, stream>>>(h, dinv, out, N);

    // 6: edge scatter (8 edges per 256-thread block)
    gcn_edge_scatter<<<(E + 7) / 8, B, 0, stream>>>(ei, h, dinv, out, E);

    // 7: bias + ReLU
    gcn_finalize<<<(N * 32 + B - 1) / B, B, 0, stream>>>(out, b, N * 32);
}